// DSNTDoubleLossNew_20083267076463
// MI455X (gfx1250) — compile-verified
//
#include <hip/hip_runtime.h>
#include <math.h>

// DSNT double loss for B=32, C=16, H=W=256.
//   Per heatmap: softmax-expected coords of `input`, argmax coords of `target`,
//   Euclidean distance; sum over 512 heatmaps / 32.
// Memory-bound: 256 MB read once -> ~11 us floor @ 23.3 TB/s. Single pass,
// non-temporal b128 loads, fixed-column thread mapping so x-weights are applied
// once after the loop and y-weights cost one FMA per 4 elements.
// exp() without max-subtraction is exact-equivalent (softmax shift invariance)
// and numerically safe for N(0,1) inputs (|x| << 80).

typedef float v2f __attribute__((ext_vector_type(2)));
typedef float v4f __attribute__((ext_vector_type(4)));
typedef float v8f __attribute__((ext_vector_type(8)));

#if defined(__has_builtin)
#if __has_builtin(__builtin_amdgcn_wmma_f32_16x16x4_f32)
#define HAVE_WMMA_F32_16X16X4 1
#endif
#endif

// Full-wave (32 lane) f32 sum. Primary path: one V_WMMA_F32_16X16X4_F32 with
// A = lane partials (A[m][0]=v[m], A[m][2]=v[m+16], k=1,3 zero), B = ones.
// Then D[m][n] = v[m]+v[m+16] for every n; lanes 0-15 hold D rows 0-7 in their
// 8 C/D VGPRs, lanes 16-31 hold rows 8-15, so sum-of-8-VGPRs + shfl_xor(16)
// gives the exact f32 total in all lanes. Requires full EXEC: called
// unconditionally by every wave.
__device__ __forceinline__ float wave_sum32(float v) {
#ifdef HAVE_WMMA_F32_16X16X4
  v2f a; a.x = v;    a.y = 0.0f;   // A-matrix 16x4: vgpr0 -> k=0 (lanes 0-15) / k=2 (lanes 16-31)
  v2f b; b.x = 1.0f; b.y = 1.0f;   // B = ones(4x16)
  v8f c = {0.f, 0.f, 0.f, 0.f, 0.f, 0.f, 0.f, 0.f};
  v8f d = __builtin_amdgcn_wmma_f32_16x16x4_f32(
      /*neg_a=*/false, a, /*neg_b=*/false, b,
      /*c_mod=*/(short)0, c, /*reuse_a=*/false, /*reuse_b=*/false);
  float r = d[0] + d[1] + d[2] + d[3] + d[4] + d[5] + d[6] + d[7];
  r += __shfl_xor(r, 16, 32);
  return r;
#else
  #pragma unroll
  for (int off = 16; off; off >>= 1) v += __shfl_xor(v, off, 32);
  return v;
#endif
}

__global__ __launch_bounds__(256) void dsnt_heatmap_kernel(
    const float* __restrict__ inp, const float* __restrict__ tgt,
    float* __restrict__ ws)
{
  const int bc = blockIdx.x;                // one block per (b,c) heatmap
  const size_t base = (size_t)bc * 65536u;  // 256*256
  const float* ip = inp + base;
  const float* tp = tgt + base;

  const int t  = threadIdx.x;   // 256 threads = 8 waves
  const int cg = t & 63;        // column group: 64 groups x 4 columns = 256 cols
  const int r0 = t >> 6;        // row phase 0..3
  const int w0 = cg << 2;       // first column owned by this thread

  float s0 = 0.f, s1 = 0.f, s2 = 0.f, s3 = 0.f;  // per-column exp sums
  float sy = 0.f;                                 // y-weighted exp sum
  float mv = -INFINITY; int mi = 0;               // target argmax (val, idx)

  // 64 iterations: rows h = r0 + 4*it; wave loads are 512B contiguous.
  #pragma unroll 4
  for (int it = 0; it < 64; ++it) {
    const int h   = r0 + (it << 2);
    const int off = (h << 8) + w0;
    const v4f x = __builtin_nontemporal_load((const v4f*)(ip + off));
    const v4f g = __builtin_nontemporal_load((const v4f*)(tp + off));

    const float e0 = __expf(x.x), e1 = __expf(x.y),
                e2 = __expf(x.z), e3 = __expf(x.w);
    s0 += e0; s1 += e1; s2 += e2; s3 += e3;
    const float yh = (2.0f * (float)h - 255.0f) * (1.0f / 256.0f);
    sy = fmaf(e0 + e1 + e2 + e3, yh, sy);

    // strictly-greater keeps the earliest index within this thread
    if (g.x > mv) { mv = g.x; mi = off;     }
    if (g.y > mv) { mv = g.y; mi = off + 1; }
    if (g.z > mv) { mv = g.z; mi = off + 2; }
    if (g.w > mv) { mv = g.w; mi = off + 3; }
  }

  // apply constant x-weights once (xs[w] = (2w-255)/256)
  const float xA = (2.0f * (float)(w0    ) - 255.0f) * (1.0f / 256.0f);
  const float xB = (2.0f * (float)(w0 + 1) - 255.0f) * (1.0f / 256.0f);
  const float xC = (2.0f * (float)(w0 + 2) - 255.0f) * (1.0f / 256.0f);
  const float xD = (2.0f * (float)(w0 + 3) - 255.0f) * (1.0f / 256.0f);
  const float sxp = xA * s0 + xB * s1 + xC * s2 + xD * s3;
  const float st  = s0 + s1 + s2 + s3;

  // wave-level reductions (WMMA path), unconditional -> EXEC all ones
  const float S  = wave_sum32(st);
  const float SX = wave_sum32(sxp);
  const float SY = wave_sum32(sy);

  // wave-level argmax with deterministic lowest-index tie-break
  #pragma unroll
  for (int off = 16; off; off >>= 1) {
    const float ov = __shfl_xor(mv, off, 32);
    const int   oi = __shfl_xor(mi, off, 32);
    if (ov > mv || (ov == mv && oi < mi)) { mv = ov; mi = oi; }
  }

  __shared__ float wS[8], wSX[8], wSY[8], wM[8];
  __shared__ int   wI[8];
  const int wid = t >> 5;
  if ((t & 31) == 0) { wS[wid] = S; wSX[wid] = SX; wSY[wid] = SY;
                       wM[wid] = mv; wI[wid] = mi; }
  __syncthreads();

  if (t == 0) {
    float aS = 0.f, aX = 0.f, aY = 0.f, bm = -INFINITY; int bi = 0;
    #pragma unroll
    for (int i = 0; i < 8; ++i) {
      aS += wS[i]; aX += wSX[i]; aY += wSY[i];
      if (wM[i] > bm || (wM[i] == bm && wI[i] < bi)) { bm = wM[i]; bi = wI[i]; }
    }
    const float px = aX / aS;
    const float py = aY / aS;
    const float tx = (2.0f * (float)(bi & 255) - 255.0f) * (1.0f / 256.0f);
    const float ty = (2.0f * (float)(bi >> 8)  - 255.0f) * (1.0f / 256.0f);
    const float dx = tx - px, dy = ty - py;
    ws[bc] = sqrtf(dx * dx + dy * dy);
  }
}

__global__ __launch_bounds__(256) void dsnt_finalize_kernel(
    const float* __restrict__ ws, float* __restrict__ out)
{
  __shared__ float red[8];
  const int t = threadIdx.x;
  float v = ws[t] + ws[t + 256];
  #pragma unroll
  for (int off = 16; off; off >>= 1) v += __shfl_xor(v, off, 32);
  if ((t & 31) == 0) red[t >> 5] = v;
  __syncthreads();
  if (t == 0) {
    float s = 0.f;
    #pragma unroll
    for (int i = 0; i < 8; ++i) s += red[i];
    out[0] = s * (1.0f / 32.0f);   // divide by B
  }
}

extern "C" void kernel_launch(void* const* d_in, const int* in_sizes, int n_in,
                              void* d_out, int out_size, void* d_ws, size_t ws_size,
                              hipStream_t stream)
{
  const float* inp = (const float*)d_in[0];  // [32,16,256,256] f32
  const float* tgt = (const float*)d_in[1];  // [32,16,256,256] f32
  float* ws  = (float*)d_ws;                 // 512 per-heatmap distances
  float* out = (float*)d_out;                // 1 f32

  dsnt_heatmap_kernel<<<512, 256, 0, stream>>>(inp, tgt, ws);
  dsnt_finalize_kernel<<<1, 256, 0, stream>>>(ws, out);
}